// Encoder_70781061038947
// MI455X (gfx1250) — compile-verified
//
#include <hip/hip_runtime.h>
#include <math.h>

// Encoder: per-row degree-16 complex polynomial from sign-selected roots
// (Vieta recurrence) + L2 normalize. Output treated as complex128 (B,17)
// flattened to doubles: out[row*34 + 2*d] = Re(c_d), [.. + 1] = Im(c_d).
//
// Memory-bound kernel (~87 MB traffic -> ~3.8us floor @ 23.3 TB/s).
// fp64 compute (~0.33G FMA) done in registers with the triangular structure.
// Output streamed via CDNA5 async LDS->global stores (ASYNCcnt path).

#define KDIM   16
#define NC     17                 // K+1 coefficients
#define BLOCK  256
#define WAVESZ 32
#define NWAVES (BLOCK / WAVESZ)
#define ROW_D2 NC                 // 17 double2 per row (272 B)
#define WAVE_D2 (WAVESZ * ROW_D2) // 544 double2 per wave (8704 B)

// ---------------------------------------------------------------------------
// Prologue: build the 16-entry root table in fp64 once per launch.
// tbl[k] = { R*cos(phi_k), R*sin(phi_k), cos(phi_k)/R, sin(phi_k)/R }
// ---------------------------------------------------------------------------
__global__ void build_root_table(const float* __restrict__ sv,
                                 double4* __restrict__ tbl) {
    int k = threadIdx.x;
    if (k < KDIM) {
        double phi = (double)sv[k];
        double R   = sqrt(1.0 + sin(M_PI / (double)KDIM));
        double c   = cos(phi);
        double s   = sin(phi);
        tbl[k] = make_double4(R * c, R * s, c / R, s / R);
    }
}

// ---------------------------------------------------------------------------
// Main kernel: one thread per row.
// ---------------------------------------------------------------------------
__global__ __launch_bounds__(BLOCK) void encoder_poly(
    const float* __restrict__ x,
    const double4* __restrict__ tbl,
    double* __restrict__ out,
    int B) {
    __shared__ double2 lds[NWAVES * WAVE_D2];  // 69632 B (<< 320 KB/WGP)

    const int tid  = threadIdx.x;
    const int lane = tid & (WAVESZ - 1);
    const int wave = tid >> 5;
    const long long row = (long long)blockIdx.x * BLOCK + tid;

    double re[NC], im[NC];
    double scale = 0.0;

    if (row < B) {
        // Coalesced row load: 4 x global_load_b128.
        const float4* xp = reinterpret_cast<const float4*>(x + row * KDIM);
        float4 q0 = xp[0], q1 = xp[1], q2 = xp[2], q3 = xp[3];
        const float xv[KDIM] = {q0.x, q0.y, q0.z, q0.w,
                                q1.x, q1.y, q1.z, q1.w,
                                q2.x, q2.y, q2.z, q2.w,
                                q3.x, q3.y, q3.z, q3.w};

        re[0] = 1.0; im[0] = 0.0;
#pragma unroll
        for (int k = 0; k < KDIM; ++k) {
            const double4 t = tbl[k];          // uniform -> scalar loads
            const bool  pos = xv[k] > 0.0f;
            const double rr = pos ? t.x : t.z; // Re(r_k)
            const double ri = pos ? t.y : t.w; // Im(r_k)
            re[k + 1] = 0.0; im[k + 1] = 0.0;
            // Triangular update: c[d] -= r * c[d-1], descending d.
#pragma unroll
            for (int d = k + 1; d >= 1; --d) {
                const double sre = re[d - 1], sim = im[d - 1];
                re[d] -= rr * sre - ri * sim;
                im[d] -= rr * sim + ri * sre;
            }
        }

        double s2 = 0.0;
#pragma unroll
        for (int d = 0; d < NC; ++d) s2 += re[d] * re[d] + im[d] * im[d];
        scale = sqrt((double)NC / s2);  // sqrt(K+1)/||c||
    }

    // Each wave owns 32 consecutive rows = 8704 contiguous output bytes.
    const long long waveRow0 = (long long)blockIdx.x * BLOCK + (long long)wave * WAVESZ;
    const bool fullWave = (waveRow0 + WAVESZ) <= (long long)B;

    if (fullWave) {
        // Stage this lane's 272 B row into LDS (ds_store_b128 x17).
        double2* myrow = &lds[wave * WAVE_D2 + lane * ROW_D2];
#pragma unroll
        for (int d = 0; d < NC; ++d)
            myrow[d] = make_double2(re[d] * scale, im[d] * scale);

        // LDS writes must land before the async engine reads them.
        asm volatile("s_wait_dscnt 0x0" ::: "memory");

        const unsigned long long gbase =
            (unsigned long long)(out + waveRow0 * 2 * NC);
        const unsigned lbase = (unsigned)(size_t)&lds[wave * WAVE_D2];

        // 17 async LDS->global b128 stores: 512 coalesced bytes each.
#pragma unroll
        for (int i = 0; i < NC; ++i) {
            const unsigned voff = (unsigned)i * 512u + (unsigned)lane * 16u;
            unsigned long long ga = gbase + voff;
            unsigned la           = lbase + voff;
            asm volatile("global_store_async_from_lds_b128 %0, %1, off"
                         :: "v"(ga), "v"(la)
                         : "memory");
        }
        asm volatile("s_wait_asynccnt 0x0" ::: "memory");
    } else if (row < B) {
        // Tail (partial wave): direct per-row stores.
        double2* orow = reinterpret_cast<double2*>(out + row * 2 * NC);
#pragma unroll
        for (int d = 0; d < NC; ++d)
            orow[d] = make_double2(re[d] * scale, im[d] * scale);
    }
}

// ---------------------------------------------------------------------------
extern "C" void kernel_launch(void* const* d_in, const int* in_sizes, int n_in,
                              void* d_out, int out_size, void* d_ws, size_t ws_size,
                              hipStream_t stream) {
    const float* x  = (const float*)d_in[0];   // (B, 16) fp32
    const float* sv = (const float*)d_in[1];   // (16,)   fp32 phases
    double*   out   = (double*)d_out;          // (B, 17) complex128 interleaved
    double4*  tbl   = (double4*)d_ws;          // 16 x 32 B root table

    const int B = in_sizes[0] / KDIM;

    build_root_table<<<1, WAVESZ, 0, stream>>>(sv, tbl);

    const int blocks = (B + BLOCK - 1) / BLOCK;
    encoder_poly<<<blocks, BLOCK, 0, stream>>>(x, tbl, out, B);
}